// PPIMMultiModalNet_16355235463393
// MI455X (gfx1250) — compile-verified
//
#include <hip/hip_runtime.h>
#include <math.h>

typedef float v8f __attribute__((ext_vector_type(8)));
typedef float v2f __attribute__((ext_vector_type(2)));
typedef unsigned int v4u __attribute__((ext_vector_type(4)));
typedef int v8i __attribute__((ext_vector_type(8)));
typedef int v4i __attribute__((ext_vector_type(4)));

#define BATCH 1024
#define SEQ   128
#define DBERT 768
#define DFP   1024
#define DENC  256
#define DPROJH 1024
#define DPROJ 256
#define NROWS 2048   // 2*B for InfoNCE

#if defined(__has_builtin)
#if __has_builtin(__builtin_amdgcn_tensor_load_to_lds) && \
    __has_builtin(__builtin_amdgcn_s_wait_tensorcnt)
#define HAVE_TDM 1
#endif
#endif

__device__ __forceinline__ v8f wmma_f32_k4(v2f a, v2f b, v8f c) {
  // D = A(16x4,f32) * B(4x16,f32) + C(16x16,f32)
  return __builtin_amdgcn_wmma_f32_16x16x4_f32(false, a, false, b, (short)0, c, false, false);
}

__device__ __forceinline__ float elu_f(float x) { return x > 0.f ? x : expm1f(x); }
__device__ __forceinline__ float gelu_f(float x) {
  return 0.5f * x * (1.f + erff(x * 0.70710678118654752f));
}

// ---------------------------------------------------------------------------
// Stage A: masked mean pool of (first+last)/2 over sequence -> pooled[B,768]
// 192 threads/block, one float4 column chunk each (768 = 192*4) -> b128 loads.
// This stage reads ~805MB and sets the kernel's HBM roofline floor.
// ---------------------------------------------------------------------------
__global__ void pool_kernel(const float* __restrict__ fh, const float* __restrict__ lh,
                            const float* __restrict__ mask, float* __restrict__ pooled) {
  const int b = blockIdx.x;
  const int tid = threadIdx.x;           // 0..191
  const float* mrow = mask + (size_t)b * SEQ;
  float msum = 0.f;
  for (int s = 0; s < SEQ; ++s) msum += mrow[s];
  const float inv = 0.5f / msum;
  float ax = 0.f, ay = 0.f, az = 0.f, aw = 0.f;
  const size_t base = (size_t)b * SEQ * DBERT;
  for (int s = 0; s < SEQ; ++s) {
    const float mk = mrow[s];
    const float4 f = ((const float4*)(fh + base + (size_t)s * DBERT))[tid];
    const float4 l = ((const float4*)(lh + base + (size_t)s * DBERT))[tid];
    ax += (f.x + l.x) * mk; ay += (f.y + l.y) * mk;
    az += (f.z + l.z) * mk; aw += (f.w + l.w) * mk;
  }
  float4 o; o.x = ax * inv; o.y = ay * inv; o.z = az * inv; o.w = aw * inv;
  ((float4*)(pooled + (size_t)b * DBERT))[tid] = o;
}

// ---------------------------------------------------------------------------
// Stage B: C[M,N] = act(A[M,K] @ W[K,N] + bias[N]); act 0=none 1=ELU 2=GELU
// Block 256 thr = 8 waves; wave computes 16x64; block tile 128(M)x64(N).
// grid = dim3(N/64, M/128). A/W stay L2-resident (192MB L2 >> working set).
// ---------------------------------------------------------------------------
__global__ void gemm_bias_act(const float* __restrict__ A, const float* __restrict__ W,
                              const float* __restrict__ bias, float* __restrict__ C,
                              int M, int N, int K, int act) {
  const int tid  = threadIdx.x;
  const int lane = tid & 31;
  const int wv   = tid >> 5;            // wave 0..7
  const int half = lane >> 4;           // K sub-split per ISA A/B layout
  const int r    = lane & 15;
  const int tm   = blockIdx.y * 8 + wv; // 16-row tile
  const int tn0  = blockIdx.x * 4;      // first 16-col tile of 4

  v8f acc[4];
  acc[0] = v8f{}; acc[1] = v8f{}; acc[2] = v8f{}; acc[3] = v8f{};

  const float* arow = A + (size_t)(tm * 16 + r) * K;
  const int c0 = tn0 * 16 + r;
  for (int k0 = 0; k0 < K; k0 += 4) {
    v2f a = *(const v2f*)(arow + k0 + 2 * half);        // {K=k0+2h, k0+2h+1}
    const float* wr0 = W + (size_t)(k0 + 2 * half) * N; // B rows k0+2h, +1
    const float* wr1 = wr0 + N;
    v2f b0, b1, b2, b3;
    b0.x = wr0[c0];      b0.y = wr1[c0];
    b1.x = wr0[c0 + 16]; b1.y = wr1[c0 + 16];
    b2.x = wr0[c0 + 32]; b2.y = wr1[c0 + 32];
    b3.x = wr0[c0 + 48]; b3.y = wr1[c0 + 48];
    acc[0] = wmma_f32_k4(a, b0, acc[0]);
    acc[1] = wmma_f32_k4(a, b1, acc[1]);
    acc[2] = wmma_f32_k4(a, b2, acc[2]);
    acc[3] = wmma_f32_k4(a, b3, acc[3]);
  }
#pragma unroll
  for (int t = 0; t < 4; ++t) {
    const int col = (tn0 + t) * 16 + r;
    const float bb = bias[col];
#pragma unroll
    for (int i = 0; i < 8; ++i) {
      const int ro = tm * 16 + half * 8 + i;   // C layout: VGPR i -> M=i / 8+i
      float v = acc[t][i] + bb;
      if (act == 1)      v = elu_f(v);
      else if (act == 2) v = gelu_f(v);
      C[(size_t)ro * N + col] = v;
    }
  }
}

// ---------------------------------------------------------------------------
// Stage C: per-sample 4x4 cosine graph -> graph[B,16]. One wave per sample.
// ---------------------------------------------------------------------------
__global__ void graph_kernel(const float* __restrict__ feats, float* __restrict__ graph) {
  const int gw   = (blockIdx.x * blockDim.x + threadIdx.x) >> 5; // sample b
  const int lane = threadIdx.x & 31;
  float dot[4][4];
#pragma unroll
  for (int i = 0; i < 4; ++i)
#pragma unroll
    for (int j = 0; j < 4; ++j) dot[i][j] = 0.f;
  for (int t = 0; t < 8; ++t) {
    const int d = lane + t * 32;
    float v[4];
#pragma unroll
    for (int m = 0; m < 4; ++m)
      v[m] = feats[((size_t)m * BATCH + gw) * DPROJ + d];
#pragma unroll
    for (int i = 0; i < 4; ++i)
#pragma unroll
      for (int j = 0; j < 4; ++j) dot[i][j] += v[i] * v[j];
  }
#pragma unroll
  for (int i = 0; i < 4; ++i)
#pragma unroll
    for (int j = 0; j < 4; ++j)
      for (int off = 1; off < 32; off <<= 1)
        dot[i][j] += __shfl_xor(dot[i][j], off, 32);
  if (lane == 0) {
    float n[4];
#pragma unroll
    for (int i = 0; i < 4; ++i) n[i] = fmaxf(sqrtf(dot[i][i]), 1e-8f);
#pragma unroll
    for (int i = 0; i < 4; ++i) {
      float sim[4], mx = -1e30f;
#pragma unroll
      for (int j = 0; j < 4; ++j) { sim[j] = dot[i][j] / (n[i] * n[j]); mx = fmaxf(mx, sim[j]); }
      float es = 0.f, e[4];
#pragma unroll
      for (int j = 0; j < 4; ++j) { e[j] = __expf(sim[j] - mx); es += e[j]; }
      float rev[4], rn = 0.f;
#pragma unroll
      for (int j = 0; j < 4; ++j) { rev[j] = sim[j] * (e[j] / es); rn += rev[j] * rev[j]; }
      rn = fmaxf(sqrtf(rn), 1e-12f);
#pragma unroll
      for (int j = 0; j < 4; ++j) graph[(size_t)gw * 16 + i * 4 + j] = rev[j] / rn;
    }
  }
}

// ---------------------------------------------------------------------------
// Stage C2: L2-normalize each feature row (eps 1e-12). One wave per row.
// feats/out are [4*1024, 256] contiguous.
// ---------------------------------------------------------------------------
__global__ void rownorm_kernel(const float* __restrict__ feats, float* __restrict__ out) {
  const int gw   = (blockIdx.x * blockDim.x + threadIdx.x) >> 5;
  const int lane = threadIdx.x & 31;
  const float* src = feats + (size_t)gw * DPROJ;
  float ss = 0.f;
  for (int d = lane; d < DPROJ; d += 32) { float v = src[d]; ss += v * v; }
  for (int off = 1; off < 32; off <<= 1) ss += __shfl_xor(ss, off, 32);
  const float inv = 1.f / fmaxf(sqrtf(ss), 1e-12f);
  float* dst = out + (size_t)gw * DPROJ;
  for (int d = lane; d < DPROJ; d += 32) dst[d] = src[d] * inv;
}

// ---------------------------------------------------------------------------
// Stage D: fused InfoNCE. grid = (128 row-tiles, 6 pairs), block = 256 (8 waves).
// The 16x256 f32 A tile is one contiguous 16KB block of fnorm (row tiles never
// straddle the 1024-row modality boundary), so wave 0 stages it with a single
// TDM tensor_load_to_lds: 1D descriptor, data_size=4B, tensor_dim0=tile_dim0=
// 4096 elements, LDS padding 4 DWORDs per 256 DWORDs -> the [16][260]
// bank-conflict-free layout is produced by the DMA engine itself.
// rows_out[p*2048 + r] = logsumexp_offdiag(row) - positive_logit.
// ---------------------------------------------------------------------------
__global__ void pair_nce_kernel(const float* __restrict__ fnorm,
                                const float* __restrict__ graph,
                                float* __restrict__ rows_out) {
  const int pairs_i[6] = {0, 0, 0, 1, 1, 2};
  const int pairs_j[6] = {1, 2, 3, 2, 3, 3};
  const int p    = blockIdx.y;
  const int tile = blockIdx.x;           // 0..127 (16 rows each)
  const int pi = pairs_i[p], pj = pairs_j[p];
  const int tid  = threadIdx.x;
  const int lane = tid & 31;
  const int wv   = tid >> 5;
  const int half = lane >> 4;
  const int r    = lane & 15;

  __shared__ float As[16][260];          // row tile, padded stride (TDM pads)
  __shared__ float scale_sh[16];
  __shared__ float pos_sh[16];
  __shared__ float partial[8][16];

  {
    const int row0 = tile * 16;
    const int m = (row0 < BATCH) ? pi : pj;
    const float* src = fnorm + ((size_t)m * BATCH + (row0 & (BATCH - 1))) * DPROJ;
#ifdef HAVE_TDM
    if (wv == 0) {
      const unsigned long long ga = (unsigned long long)(uintptr_t)src;
      const unsigned int lds = (unsigned int)(uintptr_t)&As[0][0];
      v4u g0;
      g0[0] = 1u;                                   // count=1 (valid descriptor)
      g0[1] = lds;                                  // lds_addr
      g0[2] = (unsigned int)ga;                     // global_addr[31:0]
      g0[3] = (unsigned int)(ga >> 32) | (2u << 30);// global_addr[56:32] | type=2
      v8i g1;
      // data_size=4B (2<<16), pad_enable (1<<20), pad_interval=256dw (7<<22),
      // pad_amount=4dw (3<<25), workgroup_mask=0
      g1[0] = (int)((2u << 16) | (1u << 20) | (7u << 22) | (3u << 25));
      g1[1] = (int)(4096u << 16);                   // tensor_dim0[15:0]=4096
      g1[2] = (int)(1u << 16);                      // tensor_dim0 hi=0 | tensor_dim1=1
      g1[3] = (int)(4096u << 16);                   // tensor_dim1 hi=0 | tile_dim0=4096
      g1[4] = 0;                                    // tile_dim1=0, tile_dim2=0 (unused)
      g1[5] = 4096;                                 // tensor_dim0_stride
      g1[6] = 0; g1[7] = 0;
      v4i gz4; gz4[0] = 0; gz4[1] = 0; gz4[2] = 0; gz4[3] = 0;
      v8i gz8;
#pragma unroll
      for (int q = 0; q < 8; ++q) gz8[q] = 0;
      __builtin_amdgcn_tensor_load_to_lds(g0, g1, gz4, gz4, gz8, 0);
      __builtin_amdgcn_s_wait_tensorcnt(0);         // TENSORcnt is per-wave
    }
#else
    const int rr = tid >> 4;             // row 0..15
    const int cb = (tid & 15) * 16;      // 16 floats per thread
#pragma unroll
    for (int q = 0; q < 16; ++q) As[rr][cb + q] = src[(size_t)rr * DPROJ + cb + q];
#endif
  }
  if (tid < 16) {
    const int grow = tile * 16 + tid;
    scale_sh[tid] = graph[(size_t)(grow & (BATCH - 1)) * 16 + pi * 4 + pj];
  }
  __syncthreads();                       // publish TDM-written LDS to all waves

  float sumexp[8];
#pragma unroll
  for (int i = 0; i < 8; ++i) sumexp[i] = 0.f;

  for (int q = 0; q < 16; ++q) {         // 16 column tiles per wave (128 total)
    const int ct   = wv * 16 + q;
    const int gcol = ct * 16 + r;
    const int mc   = (gcol < BATCH) ? pi : pj;
    const float* cb = fnorm + ((size_t)mc * BATCH + (gcol & (BATCH - 1))) * DPROJ;
    v8f acc = v8f{};
    for (int k0 = 0; k0 < DPROJ; k0 += 4) {
      v2f a = *(const v2f*)&As[r][k0 + 2 * half];
      v2f b = *(const v2f*)(cb + k0 + 2 * half);     // B = fn^T frag
      acc = wmma_f32_k4(a, b, acc);
    }
#pragma unroll
    for (int i = 0; i < 8; ++i) {
      const int rl   = half * 8 + i;
      const int grow = tile * 16 + rl;
      const float val = scale_sh[rl] * acc[i];
      if (gcol == ((grow + BATCH) & (NROWS - 1))) pos_sh[rl] = val;  // unique writer
      if (gcol != grow) sumexp[i] += __expf(val);                    // mask diagonal
    }
  }
#pragma unroll
  for (int i = 0; i < 8; ++i) {          // reduce across 16 lanes per half
    float v = sumexp[i];
    for (int off = 1; off < 16; off <<= 1) v += __shfl_xor(v, off, 32);
    if (r == 0) partial[wv][half * 8 + i] = v;
  }
  __syncthreads();
  if (tid < 16) {
    float s = 0.f;
#pragma unroll
    for (int w = 0; w < 8; ++w) s += partial[w][tid];   // fixed-order, deterministic
    rows_out[(size_t)p * NROWS + tile * 16 + tid] = logf(s) - pos_sh[tid];
  }
}

// ---------------------------------------------------------------------------
// Stage E: pred[b] = sum_k mw[b,k] * <feats[k][b,:], pfw[k*256:]> + pfb
// ---------------------------------------------------------------------------
__global__ void pred_kernel(const float* __restrict__ feats, const float* __restrict__ graph,
                            const float* __restrict__ pfw, const float* __restrict__ pfb,
                            float* __restrict__ out) {
  const int gw   = (blockIdx.x * blockDim.x + threadIdx.x) >> 5; // sample b
  const int lane = threadIdx.x & 31;
  float s = 0.f;
#pragma unroll
  for (int k = 0; k < 4; ++k) {
    const float mw = graph[(size_t)gw * 16 + k * 5];    // diagonal entry
    const float* f = feats + ((size_t)k * BATCH + gw) * DPROJ;
    const float* w = pfw + k * DPROJ;
    for (int d = lane; d < DPROJ; d += 32) s += mw * f[d] * w[d];
  }
  for (int off = 1; off < 32; off <<= 1) s += __shfl_xor(s, off, 32);
  if (lane == 0) out[gw] = s + pfb[0];
}

// ---------------------------------------------------------------------------
// Stage F: clr = sum(rows) / (6*2048), deterministic tree reduce.
// ---------------------------------------------------------------------------
__global__ void clr_kernel(const float* __restrict__ rows, float* __restrict__ out) {
  __shared__ float sh[256];
  const int tid = threadIdx.x;
  float s = 0.f;
  for (int i = tid; i < 6 * NROWS; i += 256) s += rows[i];
  sh[tid] = s;
  __syncthreads();
  for (int off = 128; off > 0; off >>= 1) {
    if (tid < off) sh[tid] += sh[tid + off];
    __syncthreads();
  }
  if (tid == 0) out[0] = sh[0] / (6.f * (float)NROWS);
}

// ---------------------------------------------------------------------------
// Host launcher
// ---------------------------------------------------------------------------
struct Mlp { const float *w1, *b1, *w2, *b2; };

extern "C" void kernel_launch(void* const* d_in, const int* in_sizes, int n_in,
                              void* d_out, int out_size, void* d_ws, size_t ws_size,
                              hipStream_t stream) {
  const float* fh   = (const float*)d_in[0];
  const float* lh   = (const float*)d_in[1];
  const float* mask = (const float*)d_in[2];
  const float* ecfp = (const float*)d_in[3];
  const float* hash = (const float*)d_in[4];
  const float* rdkf = (const float*)d_in[5];

  Mlp be, ee, he, re, bp, ep, hp, rp;
  const float *pfw, *pfb;
  auto F = [&](int i) { return (const float*)d_in[i]; };
  if (n_in >= 40 && in_sizes[6] == DBERT * DENC) {
    // insertion order: be,ee,he,re,bp,ep,hp,rp each (w1,b1,w2,b2), then pf(w,b)
    Mlp* blk[8] = {&be, &ee, &he, &re, &bp, &ep, &hp, &rp};
    int idx = 6;
    for (int k = 0; k < 8; ++k) {
      blk[k]->w1 = F(idx++); blk[k]->b1 = F(idx++);
      blk[k]->w2 = F(idx++); blk[k]->b2 = F(idx++);
    }
    pfw = F(idx++); pfb = F(idx++);
  } else {
    // jax sorted-key order: be,bp,ee,ep,he,hp,pf,re,rp; leaves b1,b2,w1,w2; pf: b,w
    int idx = 6;
    auto rd = [&](Mlp* m) {
      m->b1 = F(idx++); m->b2 = F(idx++); m->w1 = F(idx++); m->w2 = F(idx++);
    };
    rd(&be); rd(&bp); rd(&ee); rd(&ep); rd(&he); rd(&hp);
    pfb = F(idx++); pfw = F(idx++);
    rd(&re); rd(&rp);
  }

  // workspace layout (floats)
  float* ws     = (float*)d_ws;
  float* pooled = ws;                                   // 1024*768
  float* tmp1   = pooled + (size_t)BATCH * DBERT;       // 1024*256
  float* enc    = tmp1 + (size_t)BATCH * DENC;          // 1024*256
  float* projh  = enc + (size_t)BATCH * DENC;           // 1024*1024
  float* feats  = projh + (size_t)BATCH * DPROJH;       // 4*1024*256
  float* fnrm   = feats + (size_t)4 * BATCH * DPROJ;    // 4*1024*256
  float* graph  = fnrm + (size_t)4 * BATCH * DPROJ;     // 1024*16
  float* rows   = graph + (size_t)BATCH * 16;           // 6*2048

  pool_kernel<<<BATCH, 192, 0, stream>>>(fh, lh, mask, pooled);

  // feats order matches reference list: 0=ecfp(ee/ep) 1=bert(be/bp) 2=hashap 3=rdk
  const float* X[4]  = {ecfp, pooled, hash, rdkf};
  const int    KX[4] = {DFP, DBERT, DFP, DFP};
  const Mlp*   E[4]  = {&ee, &be, &he, &re};
  const Mlp*   P[4]  = {&ep, &bp, &hp, &rp};
  for (int m = 0; m < 4; ++m) {
    gemm_bias_act<<<dim3(DENC / 64, BATCH / 128), 256, 0, stream>>>(
        X[m], E[m]->w1, E[m]->b1, tmp1, BATCH, DENC, KX[m], 1);
    gemm_bias_act<<<dim3(DENC / 64, BATCH / 128), 256, 0, stream>>>(
        tmp1, E[m]->w2, E[m]->b2, enc, BATCH, DENC, DENC, 1);
    gemm_bias_act<<<dim3(DPROJH / 64, BATCH / 128), 256, 0, stream>>>(
        enc, P[m]->w1, P[m]->b1, projh, BATCH, DPROJH, DENC, 2);
    gemm_bias_act<<<dim3(DPROJ / 64, BATCH / 128), 256, 0, stream>>>(
        projh, P[m]->w2, P[m]->b2, feats + (size_t)m * BATCH * DPROJ,
        BATCH, DPROJ, DPROJH, 0);
  }

  graph_kernel<<<BATCH / 8, 256, 0, stream>>>(feats, graph);
  rownorm_kernel<<<(4 * BATCH) / 8, 256, 0, stream>>>(feats, fnrm);
  pair_nce_kernel<<<dim3(NROWS / 16, 6), 256, 0, stream>>>(fnrm, graph, rows);
  pred_kernel<<<BATCH / 8, 256, 0, stream>>>(feats, graph, pfw, pfb, (float*)d_out);
  clr_kernel<<<1, 256, 0, stream>>>(rows, (float*)d_out + BATCH);
}